// LIFBank_17059610100011
// MI455X (gfx1250) — compile-verified
//
#include <hip/hip_runtime.h>

// LIF bank scan: B=16, N=2048, T=1000. 32768 independent sequential recurrences.
// Memory-bound: 131 MB read + 262 MB written, ~17us floor @ 23.3 TB/s.
// No matmul structure -> WMMA inapplicable. CDNA5 paths used:
//   * global_load_async_to_lds_b128 (ASYNCcnt engine) with a 4-buffer ring,
//     prefetch distance 3 (15 outstanding async instrs/wave) for latency hiding
//   * s_wait_asynccnt stepped drain (in-order async completion)
//   * th:TH_LOAD_NT on loads + nontemporal b128 stores (single-use streaming)

#define ALPHA_F 0.95f
#define REFRACT_F 2.0f
#define THETA_INC_F 0.35f
#define THETA_DECAY_F 0.995f

typedef float vf4 __attribute__((ext_vector_type(4))); // native vector: OK for NT store

constexpr int B_ = 16;
constexpr int N_ = 2048;
constexpr int T_ = 1000;
constexpr int ROWS = B_ * N_;          // 32768 independent rows
constexpr int THREADS = 64;            // 2 wave32 per block -> 512 blocks spread over WGPs
constexpr int WAVES = THREADS / 32;
constexpr int CHUNK = 20;              // 50 * 20 = 1000 exactly
constexpr int NCHUNK = T_ / CHUNK;     // 50
constexpr int PIECES = CHUNK * 4 / 16; // 5 x 16B pieces per row per chunk
constexpr int LSTRIDE = 28;            // 112B row stride: 16B aligned, 2-way-max bank conflicts
constexpr int NBUF = 4;                // ring buffers
constexpr int PF = 3;                  // prefetch distance (chunks)
constexpr int BUFBYTES = WAVES * 32 * LSTRIDE * 4; // bytes per ring slot

__global__ __launch_bounds__(THREADS)
void lif_scan_kernel(const float* __restrict__ u,
                     const float* __restrict__ theta_base,
                     float* __restrict__ spikes,
                     float* __restrict__ vhist)
{
    // 4 bufs x 2 waves x 32 rows x 28 floats = 28 KB static LDS
    __shared__ __attribute__((aligned(16))) float lds[NBUF][WAVES][32][LSTRIDE];

    const int tid  = threadIdx.x;
    const int lane = tid & 31;
    const int w    = tid >> 5;
    const int r0w  = blockIdx.x * THREADS + w * 32; // first row of this wave's tile
    const int r    = r0w + lane;                    // this lane's (b,n) row
    const int n    = r & (N_ - 1);

    const float tb      = theta_base[n];
    const float tb_term = tb * (1.0f - THETA_DECAY_F);

    float v = 0.0f;
    float theta = tb;
    float ref = 0.0f;

    // Precompute per-lane async-copy addresses once (coalesced lane->piece map):
    // piece index c = i*32+lane covers 32 rows x 5 sixteen-byte pieces contiguously.
    const unsigned long long ubase = (unsigned long long)(const void*)u;
    unsigned long long g0[PIECES];
    unsigned int       l0[PIECES];
#pragma unroll
    for (int i = 0; i < PIECES; ++i) {
        const int c     = i * 32 + lane;
        const int row   = c / PIECES;
        const int piece = c - row * PIECES;
        g0[i] = ubase + (unsigned long long)(r0w + row) * (unsigned long long)(T_ * 4)
                      + (unsigned long long)(piece * 16);
        l0[i] = (unsigned int)(unsigned long long)(uintptr_t)&lds[0][w][row][piece * 4];
    }

    auto issue_chunk = [&](int k, int buf) {
        const unsigned long long goff = (unsigned long long)(k * (CHUNK * 4));
        const unsigned int       loff = (unsigned int)(buf * BUFBYTES);
#pragma unroll
        for (int i = 0; i < PIECES; ++i) {
            unsigned long long g = g0[i] + goff;
            unsigned int       l = l0[i] + loff;
            asm volatile("global_load_async_to_lds_b128 %0, %1, off th:TH_LOAD_NT"
                         :: "v"(l), "v"(g) : "memory");
        }
    };

    // Prologue: fill the pipeline 3 chunks deep (15 outstanding async instrs).
#pragma unroll
    for (int k = 0; k < PF; ++k) issue_chunk(k, k);

    const size_t base = (size_t)r * (size_t)T_;
    float* sp = spikes + base;
    float* vp = vhist + base;

    for (int k = 0; k < NCHUNK; ++k) {
        const int buf = k & (NBUF - 1);
        if (k + PF < NCHUNK) {
            issue_chunk(k + PF, (k + PF) & (NBUF - 1));
            // chunks k..k+PF outstanding (<=20); <=15 left => chunk k landed
            asm volatile("s_wait_asynccnt 0xF" ::: "memory");
        } else if (k + 2 < NCHUNK) {
            asm volatile("s_wait_asynccnt 0xA" ::: "memory");
        } else if (k + 1 < NCHUNK) {
            asm volatile("s_wait_asynccnt 0x5" ::: "memory");
        } else {
            asm volatile("s_wait_asynccnt 0x0" ::: "memory");
        }

        const int t0 = k * CHUNK;
#pragma unroll
        for (int q = 0; q < CHUNK; q += 4) {
            vf4 sv, vv;
#pragma unroll
            for (int m = 0; m < 4; ++m) {
                const float ut = lds[buf][w][lane][q + m]; // ds_load (b128-vectorized)
                // gate input by refractory state (pre-decrement ref)
                const float gate = (ref > 0.0f) ? 0.0f : 1.0f;
                v = ALPHA_F * v + ut * gate;
                // hard-threshold spike vs pre-update theta
                const float s = (v >= theta) ? 1.0f : 0.0f;
                v = v - s * theta;
                ref = fmaxf(ref - 1.0f, 0.0f);
                ref = (s > 0.0f) ? REFRACT_F : ref;
                theta = theta * THETA_DECAY_F + tb_term + THETA_INC_F * s;
                sv[m] = s;
                vv[m] = v;
            }
            // single-use outputs: non-temporal 128-bit stores, 16B aligned
            __builtin_nontemporal_store(sv, (vf4*)(sp + t0 + q));
            __builtin_nontemporal_store(vv, (vf4*)(vp + t0 + q));
        }
    }
}

extern "C" void kernel_launch(void* const* d_in, const int* in_sizes, int n_in,
                              void* d_out, int out_size, void* d_ws, size_t ws_size,
                              hipStream_t stream) {
    (void)in_sizes; (void)n_in; (void)d_ws; (void)ws_size; (void)out_size;
    const float* u  = (const float*)d_in[0];
    const float* tb = (const float*)d_in[1];
    float* spikes = (float*)d_out;                       // (B,N,T) flat
    float* vhist  = spikes + (size_t)B_ * N_ * T_;       // second tuple output
    dim3 grid(ROWS / THREADS);
    dim3 block(THREADS);
    hipLaunchKernelGGL(lif_scan_kernel, grid, block, 0, stream, u, tb, spikes, vhist);
}